// SOCA_41953240547848
// MI455X (gfx1250) — compile-verified
//
#include <hip/hip_runtime.h>
#include <hip/hip_bf16.h>

// ---------------------------------------------------------------------------
// SOCA forward for MI455X (gfx1250, wave32, WMMA).
// B=16, C=256, H=W=96, M=9216.  All GEMMs in fp32 via V_WMMA_F32_16X16X4_F32
// (op is HBM-bound: ~26 GFLOP vs ~0.5 GB traffic -> keep full precision).
//
// LDS tiles are stored k-pair interleaved (v2f elements) so every WMMA A/B
// fragment is a single conflict-free ds_load_b64 into an even VGPR pair —
// no VALU re-packing before v_wmma.
// ---------------------------------------------------------------------------

typedef __attribute__((ext_vector_type(2))) float v2f;
typedef __attribute__((ext_vector_type(8))) float v8f;

#define SOCA_B   16
#define SOCA_C   256
#define SOCA_M   9216            // 96*96
#define SOCA_KC  64              // K chunk staged through LDS (32 KB tiles)

// ---------------------------------------------------------------------------
// Transposing LDS fill: dst2[(k>>1)*64 + row] = {src[k], src[k+1]}.
// Global side: float4 (global_load_b128); LDS side: contiguous b64 stores
// (same k-pair, consecutive rows across threads -> conflict-free).
// HALFK = floats of k covered per thread (KC/2).
// ---------------------------------------------------------------------------
template <int HALFK>
__device__ __forceinline__ void fill_tile_T(const float* __restrict__ srcRow,
                                            v2f* __restrict__ dst,
                                            int row, int koff) {
#pragma unroll
  for (int q = 0; q < HALFK / 4; ++q) {
    float4 v = *(const float4*)(srcRow + koff + q * 4);
    const int pr = (koff + q * 4) >> 1;          // even pair index
    dst[(pr + 0) * 64 + row] = (v2f){v.x, v.y};
    dst[(pr + 1) * 64 + row] = (v2f){v.z, v.w};
  }
}

// ---------------------------------------------------------------------------
// One KC-chunk of WMMA work for a 4-wave workgroup.  Wave w owns the 16-row
// stripe sub_i = w and 4 column subtiles.  Fragment mapping for 16x4 fp32
// A / 4x16 fp32 B (wave32): lane&15 = row/col, lane>>4 selects the k pair.
// With the interleaved layout each fragment is one ds_load_b64.
// ---------------------------------------------------------------------------
template <int KC>
__device__ __forceinline__ void wmma_chunk(const v2f* __restrict__ As2,
                                           const v2f* __restrict__ Bs2,
                                           int wave, int lane, v8f acc[4]) {
  const int fi = lane & 15;
  const int kp = lane >> 4;                       // 0 or 1: pair select
#pragma unroll
  for (int k4 = 0; k4 < KC; k4 += 4) {
    const int pb = (k4 >> 1) + kp;                // k-pair index
    const v2f a = As2[pb * 64 + wave * 16 + fi];
#pragma unroll
    for (int j = 0; j < 4; ++j) {
      const v2f bv = Bs2[pb * 64 + j * 16 + fi];
      // 8 args: (neg_a, A, neg_b, B, c_mod, C, reuse_a, reuse_b)
      acc[j] = __builtin_amdgcn_wmma_f32_16x16x4_f32(
          false, a, false, bv, (short)0, acc[j], false, false);
    }
  }
}

// ---------------------------------------------------------------------------
// Kernel 1: per-(b,c) row sums of x (for covariance centering correction).
// ---------------------------------------------------------------------------
__global__ __launch_bounds__(256)
void rowsum_kernel(const float* __restrict__ x, float* __restrict__ rowsum) {
  __shared__ float red[256];
  const int row = blockIdx.x;                       // b*C + c, 4096 rows
  const float* p = x + (size_t)row * SOCA_M;
  float s = 0.0f;
  for (int t = threadIdx.x; t < SOCA_M; t += 256) s += p[t];
  red[threadIdx.x] = s;
  __syncthreads();
  for (int off = 128; off > 0; off >>= 1) {
    if (threadIdx.x < off) red[threadIdx.x] += red[threadIdx.x + off];
    __syncthreads();
  }
  if (threadIdx.x == 0) rowsum[row] = red[0];
}

// ---------------------------------------------------------------------------
// Kernel 2: covariance pooling.
//   cov[b,i,j] = (G[i,j] - s_i*s_j/M) / M,   G = X X^T,  X = x[b] (C x M).
// Grid (B, 4, 4): each 128-thread block computes a 64x64 tile, streaming
// K=9216 in 64-wide LDS chunks (32 KB LDS).
// ---------------------------------------------------------------------------
__global__ __launch_bounds__(128)
void covpool_wmma(const float* __restrict__ x, const float* __restrict__ rowsum,
                  float* __restrict__ cov) {
  __shared__ v2f As2[(SOCA_KC / 2) * 64];
  __shared__ v2f Bs2[(SOCA_KC / 2) * 64];
  const int b  = blockIdx.x;
  const int i0 = blockIdx.y * 64;
  const int j0 = blockIdx.z * 64;
  const float* X = x + (size_t)b * SOCA_C * SOCA_M;
  const int tid  = threadIdx.x;
  const int wave = tid >> 5;
  const int lane = tid & 31;
  const int ld_row  = tid >> 1;               // 0..63  (2 threads per row)
  const int ld_koff = (tid & 1) * (SOCA_KC / 2);

  v8f acc[4];
#pragma unroll
  for (int j = 0; j < 4; ++j)
#pragma unroll
    for (int r = 0; r < 8; ++r) acc[j][r] = 0.0f;

  for (int m0 = 0; m0 < SOCA_M; m0 += SOCA_KC) {
    fill_tile_T<SOCA_KC / 2>(X + (size_t)(i0 + ld_row) * SOCA_M + m0,
                             As2, ld_row, ld_koff);
    fill_tile_T<SOCA_KC / 2>(X + (size_t)(j0 + ld_row) * SOCA_M + m0,
                             Bs2, ld_row, ld_koff);
    __syncthreads();
    wmma_chunk<SOCA_KC>(As2, Bs2, wave, lane, acc);
    __syncthreads();
  }

  // epilogue: centering correction + 1/M scale
  const float invM = 1.0f / (float)SOCA_M;
  const float* rs = rowsum + b * SOCA_C;
  const int hi = (lane >> 4) * 8;
  const int fi = lane & 15;
  float si[8];
#pragma unroll
  for (int r = 0; r < 8; ++r) si[r] = rs[i0 + wave * 16 + hi + r];
#pragma unroll
  for (int j = 0; j < 4; ++j) {
    const int colg = j0 + j * 16 + fi;
    const float sj = rs[colg];
#pragma unroll
    for (int r = 0; r < 8; ++r) {
      const int rowg = i0 + wave * 16 + hi + r;
      cov[((size_t)b << 16) + rowg * SOCA_C + colg] =
          (acc[j][r] - si[r] * sj * invM) * invM;
    }
  }
}

// ---------------------------------------------------------------------------
// Kernel 3: normA[b] = trace(cov[b])
// ---------------------------------------------------------------------------
__global__ __launch_bounds__(256)
void trace_kernel(const float* __restrict__ cov, float* __restrict__ normA) {
  __shared__ float red[256];
  const int b = blockIdx.x;
  red[threadIdx.x] = cov[((size_t)b << 16) + threadIdx.x * (SOCA_C + 1)];
  __syncthreads();
  for (int off = 128; off > 0; off >>= 1) {
    if (threadIdx.x < off) red[threadIdx.x] += red[threadIdx.x + off];
    __syncthreads();
  }
  if (threadIdx.x == 0) normA[b] = red[0];
}

// ---------------------------------------------------------------------------
// Kernel 4: A = cov/normA ; Z0 = 0.5*(3I - A)
// ---------------------------------------------------------------------------
__global__ __launch_bounds__(256)
void ns_init(const float* __restrict__ cov, const float* __restrict__ normA,
             float* __restrict__ A, float* __restrict__ Z) {
  const size_t idx = (size_t)blockIdx.x * 256 + threadIdx.x;   // 1M elements
  const int b = (int)(idx >> 16);
  const int r = (int)(idx >> 8) & 255;
  const int c = (int)idx & 255;
  const float a = cov[idx] / normA[b];
  A[idx] = a;
  Z[idx] = 0.5f * ((r == c ? 3.0f : 0.0f) - a);
}

// ---------------------------------------------------------------------------
// Kernel 5: batched 256x256x256 fp32 WMMA GEMM with fused epilogues.
//   MODE 0 (PLAIN) : D = A*B
//   MODE 1 (HALF3I): D = 0.5*(3I - A*B)
//   MODE 2 (SCALE) : D = (A*B) * sqrt(normA[b])
// ---------------------------------------------------------------------------
template <int MODE>
__global__ __launch_bounds__(128)
void bmm256_wmma(const float* __restrict__ Am, const float* __restrict__ Bm,
                 float* __restrict__ D, const float* __restrict__ normA) {
  __shared__ v2f As2[(SOCA_KC / 2) * 64];
  __shared__ v2f Bs2[(SOCA_KC / 2) * 64];
  const int b  = blockIdx.x;
  const int i0 = blockIdx.y * 64;
  const int j0 = blockIdx.z * 64;
  const size_t base = (size_t)b << 16;
  const int tid  = threadIdx.x;
  const int wave = tid >> 5;
  const int lane = tid & 31;
  const int ld_row  = tid >> 1;
  const int ld_koff = (tid & 1) * (SOCA_KC / 2);
  const int ld_pair = tid >> 2;               // 0..31 (B tile: k-pair per 4 thr)
  const int ld_joff = (tid & 3) * 16;

  v8f acc[4];
#pragma unroll
  for (int j = 0; j < 4; ++j)
#pragma unroll
    for (int r = 0; r < 8; ++r) acc[j][r] = 0.0f;

  for (int k0 = 0; k0 < SOCA_C; k0 += SOCA_KC) {
    // A tile (row-major source) -> k-pair-interleaved LDS (transposed)
    fill_tile_T<SOCA_KC / 2>(Am + base + (size_t)(i0 + ld_row) * SOCA_C + k0,
                             As2, ld_row, ld_koff);
    // B tile: read rows k=2p and 2p+1, interleave pairs in registers,
    // store 32 B of contiguous LDS per step.
    {
      const float* r0 = Bm + base + (size_t)(k0 + 2 * ld_pair) * SOCA_C +
                        j0 + ld_joff;
      const float* r1 = r0 + SOCA_C;
#pragma unroll
      for (int q = 0; q < 4; ++q) {
        float4 va = ((const float4*)r0)[q];
        float4 vb = ((const float4*)r1)[q];
        v2f* d = Bs2 + ld_pair * 64 + ld_joff + q * 4;
        d[0] = (v2f){va.x, vb.x};
        d[1] = (v2f){va.y, vb.y};
        d[2] = (v2f){va.z, vb.z};
        d[3] = (v2f){va.w, vb.w};
      }
    }
    __syncthreads();
    wmma_chunk<SOCA_KC>(As2, Bs2, wave, lane, acc);
    __syncthreads();
  }

  const int hi = (lane >> 4) * 8;
  const int fi = lane & 15;
  const float sc = (MODE == 2) ? sqrtf(normA[b]) : 1.0f;
#pragma unroll
  for (int j = 0; j < 4; ++j) {
    const int colg = j0 + j * 16 + fi;
#pragma unroll
    for (int r = 0; r < 8; ++r) {
      const int rowg = i0 + wave * 16 + hi + r;
      float v = acc[j][r];
      if (MODE == 1)      v = 0.5f * ((rowg == colg ? 3.0f : 0.0f) - v);
      else if (MODE == 2) v *= sc;
      D[base + (size_t)rowg * SOCA_C + colg] = v;
    }
  }
}

// ---------------------------------------------------------------------------
// Kernel 6: cov_sum[b,j] = mean_i S[b,i,j]   (coalesced across j)
// ---------------------------------------------------------------------------
__global__ __launch_bounds__(256)
void colmean_kernel(const float* __restrict__ S, float* __restrict__ out) {
  const int b = blockIdx.x, j = threadIdx.x;
  const float* p = S + ((size_t)b << 16) + j;
  float s = 0.0f;
  for (int i = 0; i < SOCA_C; ++i) s += p[i * SOCA_C];
  out[b * SOCA_C + j] = s * (1.0f / (float)SOCA_C);
}

// ---------------------------------------------------------------------------
// Kernel 7: gate = sigmoid(relu(cs @ w1^T + b1) @ w2^T + b2)
// ---------------------------------------------------------------------------
__global__ __launch_bounds__(256)
void fc_gate_kernel(const float* __restrict__ cs,
                    const float* __restrict__ w1, const float* __restrict__ b1,
                    const float* __restrict__ w2, const float* __restrict__ b2,
                    float* __restrict__ gate) {
  __shared__ float sh[SOCA_C];
  __shared__ float h1[32];
  const int b = blockIdx.x, t = threadIdx.x;
  sh[t] = cs[b * SOCA_C + t];
  __syncthreads();
  if (t < 32) {
    float s = b1[t];
    for (int c = 0; c < SOCA_C; ++c) s += sh[c] * w1[t * SOCA_C + c];
    h1[t] = fmaxf(s, 0.0f);
  }
  __syncthreads();
  float s = b2[t];
  for (int k = 0; k < 32; ++k) s += h1[k] * w2[t * 32 + k];
  gate[b * SOCA_C + t] = 1.0f / (1.0f + __expf(-s));
}

// ---------------------------------------------------------------------------
// Kernel 8: out = gate[b,c] * x   (float4, memory-bound)
// ---------------------------------------------------------------------------
__global__ __launch_bounds__(256)
void apply_gate_kernel(const float* __restrict__ x,
                       const float* __restrict__ gate,
                       float* __restrict__ out) {
  const size_t i4 = (size_t)blockIdx.x * 256 + threadIdx.x;  // 9437184 total
  const int b = (int)(i4 / (size_t)(SOCA_C * SOCA_M / 4));   // 589824 f4/batch
  const int c = (int)(i4 / (size_t)(SOCA_M / 4)) & 255;      // 2304 f4/channel
  const float g = gate[b * SOCA_C + c];
  float4 v = ((const float4*)x)[i4];
  v.x *= g; v.y *= g; v.z *= g; v.w *= g;
  ((float4*)out)[i4] = v;
}

// ---------------------------------------------------------------------------
extern "C" void kernel_launch(void* const* d_in, const int* in_sizes, int n_in,
                              void* d_out, int out_size, void* d_ws,
                              size_t ws_size, hipStream_t stream) {
  const float* x  = (const float*)d_in[0];
  const float* w1 = (const float*)d_in[1];
  const float* b1 = (const float*)d_in[2];
  const float* w2 = (const float*)d_in[3];
  const float* b2 = (const float*)d_in[4];
  float* out = (float*)d_out;
  float* ws  = (float*)d_ws;

  const size_t MAT = (size_t)SOCA_B * SOCA_C * SOCA_C;  // 1,048,576 floats
  float* cov    = ws;              // reused for cov_sqrt at the end
  float* A      = ws + 1 * MAT;
  float* Y0     = ws + 2 * MAT;
  float* Y1     = ws + 3 * MAT;
  float* Z0     = ws + 4 * MAT;
  float* Z1     = ws + 5 * MAT;
  float* T      = ws + 6 * MAT;
  float* rowsum = ws + 7 * MAT;          // B*C = 4096
  float* normA  = rowsum + 4096;         // 16 (padded)
  float* csum   = normA + 64;            // B*C = 4096
  float* gate   = csum + 4096;           // B*C = 4096

  const dim3 gGemm(SOCA_B, 4, 4);
  const dim3 bGemm(128);

  rowsum_kernel<<<SOCA_B * SOCA_C, 256, 0, stream>>>(x, rowsum);
  covpool_wmma<<<gGemm, bGemm, 0, stream>>>(x, rowsum, cov);
  trace_kernel<<<SOCA_B, 256, 0, stream>>>(cov, normA);
  ns_init<<<(int)(MAT / 256), 256, 0, stream>>>(cov, normA, A, Z0);

  // Y = A @ Z0 ; Z = Z0
  bmm256_wmma<0><<<gGemm, bGemm, 0, stream>>>(A, Z0, Y0, nullptr);
  float *Y = Y0, *Yn = Y1, *Z = Z0, *Zn = Z1;
  for (int it = 0; it < 3; ++it) {       // iterN=5 -> 3 middle iterations
    bmm256_wmma<1><<<gGemm, bGemm, 0, stream>>>(Z, Y, T, nullptr);   // T=.5(3I-ZY)
    bmm256_wmma<0><<<gGemm, bGemm, 0, stream>>>(Y, T, Yn, nullptr);  // Y=Y T
    bmm256_wmma<0><<<gGemm, bGemm, 0, stream>>>(T, Z, Zn, nullptr);  // Z=T Z
    float* t1 = Y; Y = Yn; Yn = t1;
    float* t2 = Z; Z = Zn; Zn = t2;
  }
  bmm256_wmma<1><<<gGemm, bGemm, 0, stream>>>(Z, Y, T, nullptr);
  bmm256_wmma<2><<<gGemm, bGemm, 0, stream>>>(Y, T, cov, normA);     // *sqrt(normA)

  colmean_kernel<<<SOCA_B, 256, 0, stream>>>(cov, csum);
  fc_gate_kernel<<<SOCA_B, 256, 0, stream>>>(csum, w1, b1, w2, b2, gate);
  apply_gate_kernel<<<(int)((size_t)SOCA_B * SOCA_C * SOCA_M / 4 / 256), 256,
                      0, stream>>>(x, gate, out);
}